// SubmodularMutualInformation_80152679678829
// MI455X (gfx1250) — compile-verified
//
#include <hip/hip_runtime.h>
#include <math.h>

#define NROWS 8192
#define DIM   1024
#define EPS_COS  1e-8f
#define EPS_NORM 1e-12f

typedef __attribute__((ext_vector_type(2))) float v2f;
typedef __attribute__((ext_vector_type(8))) float v8f;

__device__ __forceinline__ v8f wmma_f32_k4(v2f a, v2f b, v8f c) {
    // D = A(16x4) * B(4x16) + C(16x16), fp32, wave32
    return __builtin_amdgcn_wmma_f32_16x16x4_f32(
        /*neg_a=*/false, a, /*neg_b=*/false, b,
        /*c_mod=*/(short)0, c, /*reuse_a=*/false, /*reuse_b=*/false);
}

__device__ __forceinline__ float blockReduceSum(float v, float* sbuf) {
    int t = threadIdx.x;
    sbuf[t] = v; __syncthreads();
    for (int s = blockDim.x >> 1; s > 0; s >>= 1) {
        if (t < s) sbuf[t] += sbuf[t + s];
        __syncthreads();
    }
    float r = sbuf[0]; __syncthreads();
    return r;
}

// ---------------- Kernel 1: per-row norms / cd / dd -----------------------
__global__ __launch_bounds__(256) void row_stats_kernel(
    const float* __restrict__ q, const float* __restrict__ docs,
    float* __restrict__ rinv_d, float* __restrict__ rinv_c,
    float* __restrict__ cd, float* __restrict__ dd) {
    __shared__ float sbuf[256];
    int row = blockIdx.x;
    const float* drow = docs + (size_t)row * DIM;
    float sdd = 0.f, scc = 0.f, scd = 0.f;
    for (int k = threadIdx.x; k < DIM; k += 256) {
        float dv = drow[k];
        float cv = 0.5f * (q[k] + dv);
        sdd += dv * dv; scc += cv * cv; scd += cv * dv;
    }
    sdd = blockReduceSum(sdd, sbuf);
    scc = blockReduceSum(scc, sbuf);
    scd = blockReduceSum(scd, sbuf);
    if (threadIdx.x == 0) {
        float nd = fmaxf(sqrtf(sdd), EPS_COS);
        float nc = fmaxf(sqrtf(scc), EPS_COS);
        float rd = 1.f / nd, rc = 1.f / nc;
        rinv_d[row] = rd;
        rinv_c[row] = rc;
        cd[row] = scd * rd * rc;
        dd[row] = sdd * rd * rd;
    }
}

// ---------------- Kernel 2: S[d] = sum_i dn[i][d] -------------------------
__global__ __launch_bounds__(256) void colsum_kernel(
    const float* __restrict__ docs, const float* __restrict__ rinv_d,
    float* __restrict__ S) {
    int col = blockIdx.x * 256 + threadIdx.x;
    float acc = 0.f;
    for (int i = 0; i < NROWS; ++i)
        acc += docs[(size_t)i * DIM + col] * rinv_d[i];
    S[col] = acc;
}

// ---------------- Kernel 3: scores base -----------------------------------
// scores[i] = cn_i.S - cd[i]*(1 - dd[i])      (T.dn subtracted later)
__global__ __launch_bounds__(256) void score_base_kernel(
    const float* __restrict__ q, const float* __restrict__ docs,
    const float* __restrict__ S, const float* __restrict__ rinv_c,
    const float* __restrict__ cd, const float* __restrict__ dd,
    float* __restrict__ scores) {
    __shared__ float sbuf[256];
    int row = blockIdx.x;
    const float* drow = docs + (size_t)row * DIM;
    float acc = 0.f;
    for (int k = threadIdx.x; k < DIM; k += 256)
        acc += (q[k] + drow[k]) * S[k];
    acc = blockReduceSum(acc, sbuf);
    if (threadIdx.x == 0)
        scores[row] = acc * 0.5f * rinv_c[row] - cd[row] * (1.f - dd[row]);
}

// ---------------- Kernel 4: Gram M = dn^T dn (D x D), WMMA fp32 -----------
// 32x32 block per wave (2x2 register-blocked 16x16 WMMA tiles).
// Upper-triangle 32x32 blocks only (symmetry). K = NROWS.
#define NBLK_D (DIM / 32)                    // 32
#define NPAIRS32 (NBLK_D * (NBLK_D + 1) / 2) // 528 -> 66 blocks * 8 waves
__global__ __launch_bounds__(256) void gram_wmma_kernel(
    const float* __restrict__ docs, const float* __restrict__ rinv_d,
    float* __restrict__ M) {
    int wave = blockIdx.x * 8 + (threadIdx.x >> 5);
    int lane = threadIdx.x & 31;
    int half = lane >> 4;          // 0: K pair {0,1}, 1: K pair {2,3}
    int l15  = lane & 15;

    // map wave -> (br, bc) with br <= bc (wave-uniform)
    int br = 0, rem = wave;
    while (rem >= (NBLK_D - br)) { rem -= (NBLK_D - br); ++br; }
    int bc = br + rem;

    int colA0 = br * 32 + l15, colA1 = colA0 + 16;
    int colB0 = bc * 32 + l15, colB1 = colB0 + 16;

    v8f c00 = {}, c01 = {}, c10 = {}, c11 = {};
#pragma unroll 2
    for (int k0 = 0; k0 < NROWS; k0 += 4) {
        int ka = k0 + half * 2;
        float r0 = rinv_d[ka], r1 = rinv_d[ka + 1];
        const float* row0 = docs + (size_t)ka * DIM;
        const float* row1 = row0 + DIM;
        v2f a0, a1, b0, b1;
        a0.x = row0[colA0] * r0;  a0.y = row1[colA0] * r1;
        a1.x = row0[colA1] * r0;  a1.y = row1[colA1] * r1;
        b0.x = row0[colB0] * r0;  b0.y = row1[colB0] * r1;
        b1.x = row0[colB1] * r0;  b1.y = row1[colB1] * r1;
        c00 = wmma_f32_k4(a0, b0, c00);
        c01 = wmma_f32_k4(a0, b1, c01);
        c10 = wmma_f32_k4(a1, b0, c10);
        c11 = wmma_f32_k4(a1, b1, c11);
    }
    // C layout: VGPR v, lanes 0-15 -> (M=v, N=lane); lanes 16-31 -> (M=v+8)
    v8f accs[2][2] = {{c00, c01}, {c10, c11}};
#pragma unroll
    for (int ti = 0; ti < 2; ++ti) {
#pragma unroll
        for (int tj = 0; tj < 2; ++tj) {
#pragma unroll
            for (int v = 0; v < 8; ++v) {
                int mr = br * 32 + ti * 16 + v + half * 8;
                int nc = bc * 32 + tj * 16 + l15;
                float val = accs[ti][tj][v];
                M[(size_t)mr * DIM + nc] = val;
                if (br != bc) M[(size_t)nc * DIM + mr] = val;  // mirror
            }
        }
    }
}

// ---------------- Kernel 5: T = cn * M, fused scores -= T_i . dn_i --------
// 32x32 output block per wave (2x2 register-blocked), K = DIM.
#define NTB_ROWS (NROWS / 32)               // 256
#define NTB (NTB_ROWS * NBLK_D)             // 8192 -> 1024 blocks * 8 waves
__global__ __launch_bounds__(256) void tdot_wmma_kernel(
    const float* __restrict__ q, const float* __restrict__ docs,
    const float* __restrict__ M, const float* __restrict__ rinv_c,
    const float* __restrict__ rinv_d, float* __restrict__ scores) {
    int wave = blockIdx.x * 8 + (threadIdx.x >> 5);
    int lane = threadIdx.x & 31;
    int half = lane >> 4;
    int l15  = lane & 15;

    int rb = wave / NBLK_D;
    int cb = wave % NBLK_D;

    int rowA0 = rb * 32 + l15, rowA1 = rowA0 + 16;
    const float* drow0 = docs + (size_t)rowA0 * DIM;
    const float* drow1 = docs + (size_t)rowA1 * DIM;
    float rc0 = rinv_c[rowA0] * 0.5f;
    float rc1 = rinv_c[rowA1] * 0.5f;
    int colB0 = cb * 32 + l15, colB1 = colB0 + 16;

    v8f c00 = {}, c01 = {}, c10 = {}, c11 = {};
#pragma unroll 2
    for (int k0 = 0; k0 < DIM; k0 += 4) {
        int ka = k0 + half * 2;
        float qa = q[ka], qb = q[ka + 1];
        const float* mrow0 = M + (size_t)ka * DIM;
        const float* mrow1 = mrow0 + DIM;
        v2f a0, a1, b0, b1;
        a0.x = (qa + drow0[ka]) * rc0;  a0.y = (qb + drow0[ka + 1]) * rc0;
        a1.x = (qa + drow1[ka]) * rc1;  a1.y = (qb + drow1[ka + 1]) * rc1;
        b0.x = mrow0[colB0];            b0.y = mrow1[colB0];
        b1.x = mrow0[colB1];            b1.y = mrow1[colB1];
        c00 = wmma_f32_k4(a0, b0, c00);
        c01 = wmma_f32_k4(a0, b1, c01);
        c10 = wmma_f32_k4(a1, b0, c10);
        c11 = wmma_f32_k4(a1, b1, c11);
    }
    // fused epilogue: scores[mr] -= sum_nc T[mr][nc] * dn[mr][nc]
    v8f accs[2][2] = {{c00, c01}, {c10, c11}};
#pragma unroll
    for (int ti = 0; ti < 2; ++ti) {
#pragma unroll
        for (int tj = 0; tj < 2; ++tj) {
#pragma unroll
            for (int v = 0; v < 8; ++v) {
                int mr = rb * 32 + ti * 16 + v + half * 8;
                int nc = cb * 32 + tj * 16 + l15;
                float val = accs[ti][tj][v] * docs[(size_t)mr * DIM + nc] * rinv_d[mr];
                val += __shfl_xor(val, 1, 32);
                val += __shfl_xor(val, 2, 32);
                val += __shfl_xor(val, 4, 32);
                val += __shfl_xor(val, 8, 32);
                if (l15 == 0) atomicAdd(&scores[mr], -val);
            }
        }
    }
}

// ---------------- Kernel 6: masked softmax --------------------------------
__global__ __launch_bounds__(1024) void softmax_kernel(
    const float* __restrict__ scores, const int* __restrict__ mask,
    float* __restrict__ w, float* __restrict__ flags) {
    __shared__ float sbuf[1024];
    int t = threadIdx.x;
    float lmax = -INFINITY;
    int cnt = 0;
    for (int r = t; r < NROWS; r += 1024)
        if (mask[r]) { lmax = fmaxf(lmax, scores[r]); ++cnt; }
    sbuf[t] = lmax; __syncthreads();
    for (int s = 512; s > 0; s >>= 1) {
        if (t < s) sbuf[t] = fmaxf(sbuf[t], sbuf[t + s]);
        __syncthreads();
    }
    float gmax = sbuf[0]; __syncthreads();

    float lsum = 0.f;
    for (int r = t; r < NROWS; r += 1024) {
        float e = 0.f;
        if (mask[r]) e = __expf(scores[r] - gmax);
        w[r] = e;
        lsum += e;
    }
    float gsum = blockReduceSum(lsum, sbuf);
    float gcnt = blockReduceSum((float)cnt, sbuf);
    float inv = (gsum > 0.f) ? (1.f / gsum) : 0.f;
    for (int r = t; r < NROWS; r += 1024) w[r] *= inv;
    if (t == 0) flags[0] = (gcnt > 0.f) ? 1.f : 0.f;
}

// ---------------- Kernel 7: weighted average ------------------------------
__global__ __launch_bounds__(256) void wavg_kernel(
    const float* __restrict__ docs, const float* __restrict__ w,
    float* __restrict__ wa) {
    int col = blockIdx.x * 256 + threadIdx.x;
    float acc = 0.f;
    for (int i = 0; i < NROWS; ++i)
        acc += docs[(size_t)i * DIM + col] * w[i];
    wa[col] = acc;
}

// ---------------- Kernel 8: normalize / fallback --------------------------
__global__ __launch_bounds__(1024) void finalize_kernel(
    const float* __restrict__ wa, const float* __restrict__ q,
    const float* __restrict__ flags, float* __restrict__ out) {
    __shared__ float sbuf[1024];
    int col = threadIdx.x;
    float v = wa[col];
    float ss = blockReduceSum(v * v, sbuf);
    float rn = 1.f / fmaxf(sqrtf(ss), EPS_NORM);
    out[col] = (flags[0] > 0.f) ? (v * rn) : q[col];
}

extern "C" void kernel_launch(void* const* d_in, const int* in_sizes, int n_in,
                              void* d_out, int out_size, void* d_ws, size_t ws_size,
                              hipStream_t stream) {
    const float* q    = (const float*)d_in[0];   // (D,)
    const float* docs = (const float*)d_in[1];   // (N, D)
    const int*   mask = (const int*)d_in[2];     // (N,)
    float* out = (float*)d_out;                  // (D,)

    float* ws = (float*)d_ws;
    float* rinv_d = ws + 0;                      // N
    float* rinv_c = ws + NROWS;                  // N
    float* cd     = ws + 2 * NROWS;              // N
    float* dd     = ws + 3 * NROWS;              // N
    float* S      = ws + 4 * NROWS;              // D
    float* scores = ws + 4 * NROWS + DIM;        // N
    float* w      = ws + 5 * NROWS + DIM;        // N
    float* wa     = ws + 6 * NROWS + DIM;        // D
    float* flags  = ws + 6 * NROWS + 2 * DIM;    // few
    float* M      = ws + 65536;                  // D*D

    row_stats_kernel<<<NROWS, 256, 0, stream>>>(q, docs, rinv_d, rinv_c, cd, dd);
    colsum_kernel<<<DIM / 256, 256, 0, stream>>>(docs, rinv_d, S);
    score_base_kernel<<<NROWS, 256, 0, stream>>>(q, docs, S, rinv_c, cd, dd, scores);
    gram_wmma_kernel<<<NPAIRS32 / 8, 256, 0, stream>>>(docs, rinv_d, M);
    tdot_wmma_kernel<<<NTB / 8, 256, 0, stream>>>(q, docs, M, rinv_c, rinv_d, scores);
    softmax_kernel<<<1, 1024, 0, stream>>>(scores, mask, w, flags);
    wavg_kernel<<<DIM / 256, 256, 0, stream>>>(docs, w, wa);
    finalize_kernel<<<1, 1024, 0, stream>>>(wa, q, flags, out);
}